// MoNet_80633716015168
// MI455X (gfx1250) — compile-verified
//
#include <hip/hip_runtime.h>
#include <hip/hip_bf16.h>

#define N_NODES   50000
#define N_EDGES   800000
#define N_GRAPHS  256
#define HID       64
#define OUT_DIM   64
#define KKER      3
#define NLAYERS   4
#define BN_EPS    1e-5f
#define NTILES    ((KKER * OUT_DIM) / 16)   // 12

typedef __attribute__((ext_vector_type(2))) float v2f;
typedef __attribute__((ext_vector_type(8))) float v8f;

// ---------------------------------------------------------------- utilities
__global__ __launch_bounds__(256) void k_zero(float* __restrict__ p, int n) {
  int i = blockIdx.x * 256 + threadIdx.x;
  if (i < n) p[i] = 0.0f;
}

// in-degree: deg[dst[e]] += 1
__global__ __launch_bounds__(256) void k_deg(const int* __restrict__ dst,
                                             float* __restrict__ deg) {
  int e = blockIdx.x * 256 + threadIdx.x;
  if (e < N_EDGES) atomicAdd(&deg[dst[e]], 1.0f);
}

// pseudo[e] = (1/sqrt(deg[src]+1), 1/sqrt(deg[dst]+1))
__global__ __launch_bounds__(256) void k_pseudo(const int* __restrict__ src,
                                                const int* __restrict__ dst,
                                                const float* __restrict__ deg,
                                                float* __restrict__ pseudo) {
  int e = blockIdx.x * 256 + threadIdx.x;
  if (e >= N_EDGES) return;
  pseudo[2 * e]     = rsqrtf(deg[src[e]] + 1.0f);
  pseudo[2 * e + 1] = rsqrtf(deg[dst[e]] + 1.0f);
}

// h[n,c] = emb[h_idx[n], c]
__global__ __launch_bounds__(256) void k_embed(const int* __restrict__ hidx,
                                               const float* __restrict__ emb,
                                               float* __restrict__ h) {
  int i = blockIdx.x * 256 + threadIdx.x;
  if (i >= N_NODES * HID) return;
  int n = i >> 6, c = i & 63;
  h[i] = emb[hidx[n] * HID + c];
}

// ------------------------------------------------- fp32 WMMA GEMM: hk = h @ fcwT
// One wave computes the full 16(M) x 192(N) output stripe for its M tile:
// A (16x4 slice of h) is loaded once per K-step and reused by 12 independent
// V_WMMA_F32_16X16X4_F32 accumulation chains (one per 16-col N tile).
__global__ __launch_bounds__(32) void k_gemm_wmma(const float* __restrict__ h,
                                                  const float* __restrict__ fcw,
                                                  float* __restrict__ hk) {
  const int tileM = blockIdx.x;          // 0..3124
  const int lane  = threadIdx.x;         // 0..31
  const int half  = lane >> 4;           // 0: K+0/K+1,  1: K+2/K+3
  const int lid   = lane & 15;

  const float* arow = h + (tileM * 16 + lid) * HID;   // A row for this lane

  v8f acc[NTILES];
#pragma unroll
  for (int t = 0; t < NTILES; ++t) acc[t] = (v8f){};

#pragma unroll
  for (int kk = 0; kk < HID; kk += 4) {
    const int ka = kk + 2 * half;
    v2f a;
    a.x = arow[ka];  a.y = arow[ka + 1];
#pragma unroll
    for (int t = 0; t < NTILES; ++t) {
      const float* brow = fcw + (t * 16 + lid) * HID;  // B[k][n] = fcw[n*HID+k]
      v2f b;
      b.x = brow[ka];  b.y = brow[ka + 1];
      // (neg_a, A, neg_b, B, c_mod, C, reuse_a, reuse_b)
      acc[t] = __builtin_amdgcn_wmma_f32_16x16x4_f32(false, a, false, b,
                                                     (short)0, acc[t],
                                                     false, false);
    }
  }

  // D layout: VGPR i -> (M = i + 8*half, N = lid)
#pragma unroll
  for (int t = 0; t < NTILES; ++t) {
    const int nOut = t * 16 + lid;
#pragma unroll
    for (int i = 0; i < 8; ++i) {
      const int m = tileM * 16 + i + 8 * half;
      hk[m * (KKER * OUT_DIM) + nOut] = acc[t][i];
    }
  }
}

// ------------------------------------------------ per-edge Gaussian weights
// gauss[e,k] = exp(-0.5 * sum_j ((tanh(pseudo@ppwT+ppb)_j - mu_kj)*isg_kj)^2)
__global__ __launch_bounds__(256) void k_gauss(const float* __restrict__ pseudo,
                                               const float* __restrict__ ppw,  // [2,2]
                                               const float* __restrict__ ppb,  // [2]
                                               const float* __restrict__ mu,   // [3,2]
                                               const float* __restrict__ isg,  // [3,2]
                                               float* __restrict__ gauss) {    // [E,3]
  int e = blockIdx.x * 256 + threadIdx.x;
  if (e >= N_EDGES) return;
  float p0 = pseudo[2 * e], p1 = pseudo[2 * e + 1];
  float u0 = tanhf(p0 * ppw[0] + p1 * ppw[1] + ppb[0]);
  float u1 = tanhf(p0 * ppw[2] + p1 * ppw[3] + ppb[1]);
#pragma unroll
  for (int k = 0; k < KKER; ++k) {
    float d0 = (u0 - mu[2 * k])     * isg[2 * k];
    float d1 = (u1 - mu[2 * k + 1]) * isg[2 * k + 1];
    gauss[e * KKER + k] = expf(-0.5f * (d0 * d0 + d1 * d1));
  }
}

// ----------------------------------------- edge gather/scale/scatter (per layer)
// thread t -> edge e = t/64, channel o = t%64.
// agg[dst,o] += sum_k hk[src, k*64+o] * gauss[e,k]
__global__ __launch_bounds__(256) void k_edge_agg(const int* __restrict__ src,
                                                  const int* __restrict__ dst,
                                                  const float* __restrict__ gauss,
                                                  const float* __restrict__ hk,
                                                  float* __restrict__ agg) {
  long t = (long)blockIdx.x * 256 + threadIdx.x;
  int e = (int)(t >> 6);
  int o = (int)(t & 63);
  if (e >= N_EDGES) return;

  float g0 = gauss[e * KKER + 0];
  float g1 = gauss[e * KKER + 1];
  float g2 = gauss[e * KKER + 2];

  const float* hks = hk + src[e] * (KKER * OUT_DIM) + o;
  float acc = hks[0] * g0;
  acc = fmaf(hks[OUT_DIM],     g1, acc);
  acc = fmaf(hks[2 * OUT_DIM], g2, acc);
  atomicAdd(&agg[dst[e] * OUT_DIM + o], acc);
}

// ------------------------------------------------- BatchNorm stats (biased var)
// 256 blocks x 256 threads; thread = (row r in 0..3, channel c in 0..63) so each
// iteration reads 1KB contiguous. Partials combined in LDS, one atomic per
// channel into sums[0:64]=sum, sums[64:128]=sumsq.
__global__ __launch_bounds__(256) void k_bn_partial(const float* __restrict__ agg,
                                                    float* __restrict__ sums) {
  const int c = threadIdx.x & 63;
  const int r = threadIdx.x >> 6;      // 0..3
  float s = 0.0f, s2 = 0.0f;
  for (int n = blockIdx.x * 4 + r; n < N_NODES; n += 256 * 4) {
    float v = agg[n * OUT_DIM + c];
    s += v;
    s2 = fmaf(v, v, s2);
  }
  __shared__ float sh[256], sh2[256];
  sh[threadIdx.x] = s;  sh2[threadIdx.x] = s2;
  __syncthreads();
  if (r == 0) {
    s  = sh[c]  + sh[c + 64]  + sh[c + 128]  + sh[c + 192];
    s2 = sh2[c] + sh2[c + 64] + sh2[c + 128] + sh2[c + 192];
    atomicAdd(&sums[c], s);
    atomicAdd(&sums[64 + c], s2);
  }
}

__global__ __launch_bounds__(64) void k_bn_finalize(const float* __restrict__ sums,
                                                    float* __restrict__ mean,
                                                    float* __restrict__ var) {
  int c = threadIdx.x;
  float m = sums[c] * (1.0f / N_NODES);
  mean[c] = m;
  var[c]  = sums[64 + c] * (1.0f / N_NODES) - m * m;
}

// h += relu((agg-mean)*rsqrt(var+eps)*g + b)
__global__ __launch_bounds__(256) void k_bn_apply(float* __restrict__ h,
                                                  const float* __restrict__ agg,
                                                  const float* __restrict__ mean,
                                                  const float* __restrict__ var,
                                                  const float* __restrict__ g,
                                                  const float* __restrict__ b) {
  int i = blockIdx.x * 256 + threadIdx.x;
  if (i >= N_NODES * OUT_DIM) return;
  int c = i & 63;
  float hn = (agg[i] - mean[c]) * rsqrtf(var[c] + BN_EPS) * g[c] + b[c];
  h[i] += fmaxf(hn, 0.0f);
}

// ---------------------------------------------------------- graph mean readout
__global__ __launch_bounds__(256) void k_segsum(const int* __restrict__ gids,
                                                const float* __restrict__ h,
                                                float* __restrict__ hg,
                                                float* __restrict__ cnt) {
  int i = blockIdx.x * 256 + threadIdx.x;
  if (i >= N_NODES * OUT_DIM) return;
  int n = i >> 6, c = i & 63;
  int g = gids[n];
  atomicAdd(&hg[g * OUT_DIM + c], h[i]);
  if (c == 0) atomicAdd(&cnt[g], 1.0f);
}

// MLPReadout: 64 -> 32 -> 16 -> 1, one block (64 threads) per graph
__global__ __launch_bounds__(64) void k_mlp(const float* __restrict__ hg,
                                            const float* __restrict__ cnt,
                                            const float* __restrict__ w1, const float* __restrict__ b1,
                                            const float* __restrict__ w2, const float* __restrict__ b2,
                                            const float* __restrict__ w3, const float* __restrict__ b3,
                                            float* __restrict__ out) {
  const int gid = blockIdx.x;
  __shared__ float x[64], y1[32], y2[16];
  float c = fmaxf(cnt[gid], 1.0f);
  x[threadIdx.x] = hg[gid * OUT_DIM + threadIdx.x] / c;
  __syncthreads();
  if (threadIdx.x < 32) {
    float a = b1[threadIdx.x];
    for (int k = 0; k < 64; ++k) a = fmaf(w1[threadIdx.x * 64 + k], x[k], a);
    y1[threadIdx.x] = fmaxf(a, 0.0f);
  }
  __syncthreads();
  if (threadIdx.x < 16) {
    float a = b2[threadIdx.x];
    for (int k = 0; k < 32; ++k) a = fmaf(w2[threadIdx.x * 32 + k], y1[k], a);
    y2[threadIdx.x] = fmaxf(a, 0.0f);
  }
  __syncthreads();
  if (threadIdx.x == 0) {
    float a = b3[0];
    for (int k = 0; k < 16; ++k) a = fmaf(w3[k], y2[k], a);
    out[gid] = a;
  }
}

// ============================================================================
extern "C" void kernel_launch(void* const* d_in, const int* in_sizes, int n_in,
                              void* d_out, int out_size, void* d_ws, size_t ws_size,
                              hipStream_t stream) {
  (void)in_sizes; (void)n_in; (void)out_size; (void)ws_size;

  const int*   h_idx = (const int*)d_in[0];
  const int*   src   = (const int*)d_in[1];
  const int*   dst   = (const int*)d_in[2];
  const int*   gids  = (const int*)d_in[3];
  const float* emb   = (const float*)d_in[4];
  const float* fc_w  = (const float*)d_in[5];   // [L, 192, 64]
  const float* mu    = (const float*)d_in[6];   // [L, 3, 2]
  const float* isg   = (const float*)d_in[7];   // [L, 3, 2]
  const float* bng   = (const float*)d_in[8];   // [L, 64]
  const float* bnb   = (const float*)d_in[9];   // [L, 64]
  const float* ppw   = (const float*)d_in[10];  // [L, 2, 2]
  const float* ppb   = (const float*)d_in[11];  // [L, 2]
  const float* w1    = (const float*)d_in[12];
  const float* b1    = (const float*)d_in[13];
  const float* w2    = (const float*)d_in[14];
  const float* b2    = (const float*)d_in[15];
  const float* w3    = (const float*)d_in[16];
  const float* b3    = (const float*)d_in[17];

  // workspace carve-up (floats)
  float* ws     = (float*)d_ws;
  float* deg    = ws;  ws += N_NODES;
  float* pseudo = ws;  ws += 2 * N_EDGES;
  float* gauss  = ws;  ws += KKER * N_EDGES;
  float* h      = ws;  ws += N_NODES * HID;
  float* hk     = ws;  ws += N_NODES * KKER * OUT_DIM;
  float* agg    = ws;  ws += N_NODES * OUT_DIM;   // agg and sums contiguous:
  float* sums   = ws;  ws += 2 * OUT_DIM;         // zeroed together per layer
  float* mean   = ws;  ws += OUT_DIM;
  float* var    = ws;  ws += OUT_DIM;
  float* hg     = ws;  ws += N_GRAPHS * OUT_DIM;  // hg and cnt contiguous
  float* cnt    = ws;  ws += N_GRAPHS;

  const int NE_B = (N_EDGES + 255) / 256;
  const int NH_B = (N_NODES * HID + 255) / 256;

  k_zero  <<<(N_NODES + 255) / 256, 256, 0, stream>>>(deg, N_NODES);
  k_deg   <<<NE_B, 256, 0, stream>>>(dst, deg);
  k_pseudo<<<NE_B, 256, 0, stream>>>(src, dst, deg, pseudo);
  k_embed <<<NH_B, 256, 0, stream>>>(h_idx, emb, h);

  for (int l = 0; l < NLAYERS; ++l) {
    k_gemm_wmma<<<N_NODES / 16, 32, 0, stream>>>(
        h, fc_w + (size_t)l * KKER * OUT_DIM * HID, hk);
    k_gauss<<<NE_B, 256, 0, stream>>>(pseudo, ppw + l * 4, ppb + l * 2,
                                      mu + l * 6, isg + l * 6, gauss);
    k_zero<<<(N_NODES * OUT_DIM + 2 * OUT_DIM + 255) / 256, 256, 0, stream>>>(
        agg, N_NODES * OUT_DIM + 2 * OUT_DIM);
    k_edge_agg<<<(int)(((long)N_EDGES * OUT_DIM + 255) / 256), 256, 0, stream>>>(
        src, dst, gauss, hk, agg);
    k_bn_partial <<<256, 256, 0, stream>>>(agg, sums);
    k_bn_finalize<<<1, 64, 0, stream>>>(sums, mean, var);
    k_bn_apply<<<NH_B, 256, 0, stream>>>(h, agg, mean, var,
                                         bng + l * OUT_DIM, bnb + l * OUT_DIM);
  }

  k_zero<<<(N_GRAPHS * OUT_DIM + N_GRAPHS + 255) / 256, 256, 0, stream>>>(
      hg, N_GRAPHS * OUT_DIM + N_GRAPHS);
  k_segsum<<<NH_B, 256, 0, stream>>>(gids, h, hg, cnt);
  k_mlp<<<N_GRAPHS, 64, 0, stream>>>(hg, cnt, w1, b1, w2, b2, w3, b3, (float*)d_out);
}